// FraudDetectionHybridModel_65481071397769
// MI455X (gfx1250) — compile-verified
//
#include <hip/hip_runtime.h>

// ---------------------------------------------------------------------------
// Quantum-patch conv + classifier, collapsed analytically:
//   RZ phases never change |amp|^2  ->  patch_params are mathematically unused
//   feats[b, p, w] = prod_{j<=w} cos(pixel_j of 2x2 patch p)     (w = 0..3)
//   out[b]         = sigmoid(cos(sum_{p,w} feats[b,p,w] * W[p*4+w]))
//
// MI455X plan: 12.8 MB in / 16 KB out -> pure HBM-bound (~0.55us @ 23.3TB/s).
//  * Stage 16 images + all 784 weights into LDS with the CDNA5 async-to-LDS
//    path (global_load_async_to_lds_b128, ASYNCcnt) -- coalesced 16B/lane,
//    each byte of x read exactly once, no VGPR round-trip.
//  * The 784-long dot product per batch element runs on the matrix unit:
//    V_WMMA_F32_16X16X4_F32 (full fp32). A = feats*W [16 batches x 4 feats of
//    one patch], B = ones(4x16) => every D column equals the partial angle,
//    independent of B's VGPR layout; C accumulates 49 patches per wave.
//  * v_cos_f32 is a TRANS op -> co-executes with the XDL WMMA pipe.
// ---------------------------------------------------------------------------

typedef float v2f __attribute__((ext_vector_type(2)));
typedef float v8f __attribute__((ext_vector_type(8)));

#define NQ      4
#define GRID14  14
#define NP      196          // 14*14 patches
#define IMG     784          // 28*28 floats per image
#define IMG4    196          // float4s per image
#define BPB     16           // batch elements per block (one WMMA M-tile)
#define ROWP    788          // padded LDS row stride in floats: 788*4 B is
                             // 16B-aligned and 20*b mod 64 distinct for
                             // b=0..15 -> conflict-free 16-row reads
#define NWAVE   4            // 128 threads / wave32
#define PPW     49           // patches per wave (196 / 4)

__global__ __launch_bounds__(128)
void qconv_wmma_kernel(const float* __restrict__ x,
                       const float* __restrict__ wclass,
                       float* __restrict__ out,
                       int nbatch)
{
    __shared__ float simg[BPB * ROWP];        // 16 staged images (50.4 KB)
    __shared__ float swt[NP * NQ];            // 784 classifier weights
    __shared__ float spartial[NWAVE * BPB];   // per-wave partial angles

    const int t         = threadIdx.x;
    const int batchBase = blockIdx.x * BPB;

    // ---- Async-stage classifier weights: 196 x b128, LDS <- global --------
    {
        const unsigned ldsW = (unsigned)(size_t)&swt[0];   // low 32b = LDS off
        for (int idx = t; idx < IMG4; idx += 128) {        // 784 floats
            unsigned     dst = ldsW + (unsigned)(idx * 16);
            const float* src = wclass + idx * 4;
            asm volatile("global_load_async_to_lds_b128 %0, %1, off"
                         :: "v"(dst), "v"(src) : "memory");
        }
    }
    // ---- Async-stage 16 images: 3136 x b128, add-and-wrap (no divides) ----
    {
        const unsigned ldsI = (unsigned)(size_t)&simg[0];
        const float*   gx   = x + (size_t)batchBase * IMG;
        int b  = 0;          // blockDim 128 < 196 -> every thread starts in b0
        int o4 = t;
        for (int idx = t; idx < BPB * IMG4; idx += 128) {
            if (batchBase + b < nbatch) {
                unsigned     dst = ldsI + (unsigned)(b * (ROWP * 4) + o4 * 16);
                const float* src = gx + idx * 4;
                asm volatile("global_load_async_to_lds_b128 %0, %1, off"
                             :: "v"(dst), "v"(src) : "memory");
            }
            o4 += 128;
            if (o4 >= IMG4) { o4 -= IMG4; ++b; }
        }
    }
    // Drain this wave's ASYNCcnt, then make all waves' fills visible.
    asm volatile("s_wait_asynccnt 0x0" ::: "memory");
    __syncthreads();

    const int  wave = t >> 5;
    const int  lane = t & 31;
    const int  brow = lane & 15;      // which of the 16 batch rows (M)
    const bool hi   = lane >= 16;     // lanes 16..31 supply A columns K=2,3

    v8f acc = {0.f, 0.f, 0.f, 0.f, 0.f, 0.f, 0.f, 0.f};
    const v2f bones = {1.0f, 1.0f};   // B = ones(4x16): layout-independent

    // ---- Accumulate angle over this wave's 49 patches via f32 WMMA --------
    int p  = wave * PPW;              // uniform; one scalar div below only
    int pr = p / GRID14;
    int pc = p - pr * GRID14;
    for (int i = 0; i < PPW; ++i) {
        // 2x2 patch of image `brow`: rows 2*pr, 2*pr+1; cols 2*pc, 2*pc+1
        const float* base = &simg[brow * ROWP + (2 * pr) * 28 + 2 * pc];
        v2f r0 = *(const v2f*)(base);         // ds_load_b64 (conflict-free)
        v2f r1 = *(const v2f*)(base + 28);    // ds_load_b64

        float c0 = __cosf(r0.x);
        float c1 = __cosf(r0.y);
        float c2 = __cosf(r1.x);
        float c3 = __cosf(r1.y);
        float f01   = c0 * c1;
        float f012  = f01 * c2;
        float f0123 = f012 * c3;

        // classifier weights for this lane's K pair (patch-major, qubit-minor)
        v2f wv = *(const v2f*)(&swt[p * NQ + (hi ? 2 : 0)]);   // LDS broadcast

        // A[brow, K] per 16x4 f32 layout: v0 = K=0 (lo lanes) / K=2 (hi),
        //                                 v1 = K=1 (lo lanes) / K=3 (hi)
        v2f a;
        a.x = (hi ? f012  : c0 ) * wv.x;
        a.y = (hi ? f0123 : f01) * wv.y;

        // D += A x ones : every D column = partial angle for its batch row
        acc = __builtin_amdgcn_wmma_f32_16x16x4_f32(
                  /*neg_a=*/false, a, /*neg_b=*/false, bones,
                  /*c_mod=*/(short)0, acc, /*reuse_a=*/false, /*reuse_b=*/false);

        ++p;
        if (++pc == GRID14) { pc = 0; ++pr; }
    }

    // ---- Extract column 0 of D (16x16 f32 C/D layout):
    //      lane 0,  VGPR r -> D[r][0]     (batches 0..7)
    //      lane 16, VGPR r -> D[8+r][0]   (batches 8..15)
    if (lane == 0) {
        #pragma unroll
        for (int m = 0; m < 8; ++m) spartial[wave * BPB + m] = acc[m];
    }
    if (lane == 16) {
        #pragma unroll
        for (int m = 0; m < 8; ++m) spartial[wave * BPB + 8 + m] = acc[m];
    }
    __syncthreads();

    // ---- Cross-wave reduction + sigmoid(cos(angle)) -----------------------
    if (t < BPB && batchBase + t < nbatch) {
        float angle = spartial[t] + spartial[BPB + t] +
                      spartial[2 * BPB + t] + spartial[3 * BPB + t];
        float z = __cosf(angle);
        out[batchBase + t] = 1.0f / (1.0f + __expf(-z));
    }
}

extern "C" void kernel_launch(void* const* d_in, const int* in_sizes, int n_in,
                              void* d_out, int out_size, void* d_ws, size_t ws_size,
                              hipStream_t stream)
{
    const float* x  = (const float*)d_in[0];
    // d_in[1] = patch_params: RZ phases cancel in |amp|^2 -> provably unused
    const float* wc = (const float*)d_in[2];
    float*       o  = (float*)d_out;

    const int nbatch = in_sizes[0] / IMG;            // 4096
    const int blocks = (nbatch + BPB - 1) / BPB;     // 256
    qconv_wmma_kernel<<<blocks, 128, 0, stream>>>(x, wc, o, nbatch);
}